// SpatialProcessor_71571335020987
// MI455X (gfx1250) — compile-verified
//
#include <hip/hip_runtime.h>
#include <math.h>

// Problem constants (match reference)
static constexpr int GN   = 2048;   // nodes
static constexpr int INF_ = 64;     // IN_DIM
static constexpr int HID  = 128;    // hidden == out
static constexpr int HEADS = 4;
static constexpr int FH   = 32;     // per-head feature dim
static constexpr int BB   = 8;      // batch
static constexpr float NEG_SLOPE = 0.2f;

typedef float v2f __attribute__((ext_vector_type(2)));
typedef float v8f __attribute__((ext_vector_type(8)));

// ---- order-preserving float<->uint mapping for atomic min over floats ----
__device__ __forceinline__ unsigned map_f(float x) {
  unsigned u = __float_as_uint(x);
  return (u & 0x80000000u) ? ~u : (u | 0x80000000u);
}
__device__ __forceinline__ float unmap_f(unsigned m) {
  unsigned u = (m & 0x80000000u) ? (m & 0x7FFFFFFFu) : ~m;
  return __uint_as_float(u);
}

// ---------------------------------------------------------------------------
// 1) Row-normalize embedding; thread0 of block0 initializes min scratch
// ---------------------------------------------------------------------------
__global__ void normalize_emb(const float* __restrict__ emb, float* __restrict__ ne,
                              unsigned* __restrict__ minS) {
  int i = blockIdx.x;                  // row, blockDim = 128
  float v = emb[(size_t)i * HID + threadIdx.x];
  __shared__ float red[128];
  red[threadIdx.x] = v * v;
  __syncthreads();
  for (int st = 64; st > 0; st >>= 1) {
    if (threadIdx.x < st) red[threadIdx.x] += red[threadIdx.x + st];
    __syncthreads();
  }
  float rs = rsqrtf(red[0] + 1e-12f);
  ne[(size_t)i * HID + threadIdx.x] = v * rs;
  if (i == 0 && threadIdx.x == 0) *minS = 0xFFFFFFFFu;  // +inf in mapped order
}

// ---------------------------------------------------------------------------
// 2) Per-row top-2 of adj (diag treated as 0), atomic global min of 2nd-largest
// ---------------------------------------------------------------------------
__global__ void row_top2_min(const float* __restrict__ ne, unsigned* __restrict__ minS) {
  int i = blockIdx.x;                  // blockDim = 256
  __shared__ float nei[HID];
  for (int f = threadIdx.x; f < HID; f += 256) nei[f] = ne[(size_t)i * HID + f];
  __syncthreads();
  float t1 = -3.4e38f, t2 = -3.4e38f;
  for (int j = threadIdx.x; j < GN; j += 256) {
    float v = 0.0f;
    if (j != i) {
      const float* nj = ne + (size_t)j * HID;
      float acc = 0.0f;
      for (int f = 0; f < HID; ++f) acc += nei[f] * nj[f];
      v = acc;
    }
    if (v > t1) { t2 = t1; t1 = v; } else if (v > t2) { t2 = v; }
  }
  __shared__ float r1[256], r2[256];
  r1[threadIdx.x] = t1; r2[threadIdx.x] = t2;
  __syncthreads();
  for (int st = 128; st > 0; st >>= 1) {
    if (threadIdx.x < st) {
      float a1 = r1[threadIdx.x], a2 = r2[threadIdx.x];
      float b1 = r1[threadIdx.x + st], b2 = r2[threadIdx.x + st];
      r1[threadIdx.x] = fmaxf(a1, b1);
      r2[threadIdx.x] = fmaxf(fminf(a1, b1), fmaxf(a2, b2));
    }
    __syncthreads();
  }
  if (threadIdx.x == 0) atomicMin(minS, map_f(r2[0]));  // min(top1,top2)=top2
}

// ---------------------------------------------------------------------------
// 3) Build bitmask: keep = (adj >= minval) && (adj > 0.5); diag = 0 -> excluded
// ---------------------------------------------------------------------------
__global__ void build_mask(const float* __restrict__ ne, const unsigned* __restrict__ minS,
                           unsigned* __restrict__ maskb) {
  int i = blockIdx.x;                  // blockDim = 256
  __shared__ float nei[HID];
  for (int f = threadIdx.x; f < HID; f += 256) nei[f] = ne[(size_t)i * HID + f];
  __syncthreads();
  float minv = unmap_f(*minS);
  for (int j = threadIdx.x; j < GN; j += 256) {
    float v = 0.0f;
    if (j != i) {
      const float* nj = ne + (size_t)j * HID;
      float acc = 0.0f;
      for (int f = 0; f < HID; ++f) acc += nei[f] * nj[f];
      v = acc;
    }
    bool keep = (v >= minv) && (v > 0.5f);
    unsigned long long bal = __ballot(keep);      // wave32 -> low 32 bits
    if ((threadIdx.x & 31) == 0) maskb[(size_t)i * 64 + (j >> 5)] = (unsigned)bal;
  }
}

// ---------------------------------------------------------------------------
// 4) Dense fp32 GEMM via v_wmma_f32_16x16x4_f32; one wave = one 16x16 C tile
//    A[M,K] row-major, B[K,Nn] row-major, optional bias/relu.
// ---------------------------------------------------------------------------
__global__ void gemm_f32_wmma(const float* __restrict__ A, const float* __restrict__ Bm,
                              const float* __restrict__ bias, float* __restrict__ C,
                              int M, int K, int Nn, int relu) {
  int tile = blockIdx.x;
  int ntn = Nn >> 4;
  int tm = tile / ntn, tn = tile % ntn;
  int lane = threadIdx.x;
  int r = lane & 15, half = lane >> 4;
  const float* Arow = A + (size_t)(tm * 16 + r) * K;
  v8f acc = {};
  for (int k = 0; k < K; k += 4) {
    v2f a, b;
    a.x = Arow[k + 2 * half + 0];                      // A[m][k..], lane-half = K pair
    a.y = Arow[k + 2 * half + 1];
    b.x = Bm[(size_t)(k + 2 * half + 0) * Nn + tn * 16 + r];
    b.y = Bm[(size_t)(k + 2 * half + 1) * Nn + tn * 16 + r];
    acc = __builtin_amdgcn_wmma_f32_16x16x4_f32(false, a, false, b, (short)0, acc,
                                                false, false);
  }
  int col = tn * 16 + r;
  float bv = bias ? bias[col] : 0.0f;
  for (int v = 0; v < 8; ++v) {
    int m = tm * 16 + v + 8 * half;                    // C/D layout: row = v + 8*half
    float val = acc[v] + bv;
    if (relu) val = fmaxf(val, 0.0f);
    C[(size_t)m * Nn + col] = val;
  }
}

// ---------------------------------------------------------------------------
// 5) Per-head edge score projections: esrc/edst [B*H*N]
// ---------------------------------------------------------------------------
__global__ void edge_scores(const float* __restrict__ hw, const float* __restrict__ a_src,
                            const float* __restrict__ a_dst, float* __restrict__ esrc,
                            float* __restrict__ edst) {
  int idx = blockIdx.x * blockDim.x + threadIdx.x;  // (b*H + h)*N + n
  int n  = idx & (GN - 1);
  int bh = idx >> 11;
  int h  = bh & (HEADS - 1);
  int b  = bh >> 2;
  const float* hp = hw + ((size_t)(b * GN + n)) * HID + h * FH;
  float s = 0.0f, d = 0.0f;
  for (int f = 0; f < FH; ++f) {
    float hv = hp[f];
    s += hv * a_src[h * FH + f];
    d += hv * a_dst[h * FH + f];
  }
  esrc[idx] = s;
  edst[idx] = d;
}

// ---------------------------------------------------------------------------
// 6) Softmax stats per (b,h,i): masked row max m and 1/Z (0 for empty rows)
// ---------------------------------------------------------------------------
__global__ void attn_stats(const float* __restrict__ esrc, const float* __restrict__ edst,
                           const unsigned* __restrict__ maskb, float* __restrict__ mrow,
                           float* __restrict__ zinv) {
  int row = blockIdx.x;                 // (b*H + h)*N + i, blockDim = 256
  int i  = row & (GN - 1);
  int bh = row >> 11;
  const float* ed = edst + (size_t)bh * GN;
  float es = esrc[row];
  __shared__ float red[256];
  float m = -3.4e38f;
  for (int j = threadIdx.x; j < GN; j += 256) {
    unsigned w = maskb[(size_t)i * 64 + (j >> 5)];
    if ((w >> (j & 31)) & 1u) {
      float s = es + ed[j];
      s = s >= 0.0f ? s : NEG_SLOPE * s;
      m = fmaxf(m, s);
    }
  }
  red[threadIdx.x] = m;
  __syncthreads();
  for (int st = 128; st > 0; st >>= 1) {
    if (threadIdx.x < st) red[threadIdx.x] = fmaxf(red[threadIdx.x], red[threadIdx.x + st]);
    __syncthreads();
  }
  m = red[0];
  __syncthreads();
  float z = 0.0f;
  if (m > -1e38f) {
    for (int j = threadIdx.x; j < GN; j += 256) {
      unsigned w = maskb[(size_t)i * 64 + (j >> 5)];
      if ((w >> (j & 31)) & 1u) {
        float s = es + ed[j];
        s = s >= 0.0f ? s : NEG_SLOPE * s;
        z += __expf(s - m);
      }
    }
  }
  red[threadIdx.x] = z;
  __syncthreads();
  for (int st = 128; st > 0; st >>= 1) {
    if (threadIdx.x < st) red[threadIdx.x] += red[threadIdx.x + st];
    __syncthreads();
  }
  if (threadIdx.x == 0) {
    float Z = red[0];
    if (m > -1e38f && Z > 0.0f) { mrow[row] = m; zinv[row] = 1.0f / Z; }
    else                        { mrow[row] = 0.0f; zinv[row] = 0.0f; }  // isolated row
  }
}

// ---------------------------------------------------------------------------
// 7) Fused attention apply: out[i, h*32+f] = sum_j alpha[i,j] * hw[j, h*32+f]
//    alpha generated in-register, accumulated with v_wmma_f32_16x16x4_f32.
//    One wave per (b, h, 16-row tile); K-dim = neighbor index j.
// ---------------------------------------------------------------------------
__global__ void attn_apply(const float* __restrict__ hw, const float* __restrict__ esrc,
                           const float* __restrict__ edst, const float* __restrict__ mrow,
                           const float* __restrict__ zinv, const unsigned* __restrict__ maskb,
                           float* __restrict__ out, int relu) {
  int blk = blockIdx.x;                 // (b*H + h)*(N/16) + it
  int it = blk & 127;                   // N/16 = 128
  int bh = blk >> 7;
  int h  = bh & (HEADS - 1);
  int b  = bh >> 2;
  int lane = threadIdx.x;
  int r = lane & 15, half = lane >> 4;
  int i = it * 16 + r;
  int row = bh * GN + i;
  float es = esrc[row];
  float mi = mrow[row];
  float zi = zinv[row];
  const float* ed = edst + (size_t)bh * GN;
  const unsigned* mb = maskb + (size_t)i * 64;
  const float* hbase = hw + ((size_t)b * GN) * HID + h * FH;  // + j*HID + f
  v8f c0 = {}, c1 = {};
  for (int jt = 0; jt < GN; jt += 16) {
    for (int ks = 0; ks < 4; ++ks) {
      int j0 = jt + ks * 4 + 2 * half;                // even; j0,j0+1 share mask word
      unsigned w = mb[j0 >> 5];
      float s0 = es + ed[j0];       s0 = s0 >= 0.0f ? s0 : NEG_SLOPE * s0;
      float s1 = es + ed[j0 + 1];   s1 = s1 >= 0.0f ? s1 : NEG_SLOPE * s1;
      float p0 = ((w >> (j0 & 31)) & 1u)       ? __expf(s0 - mi) * zi : 0.0f;
      float p1 = ((w >> ((j0 + 1) & 31)) & 1u) ? __expf(s1 - mi) * zi : 0.0f;
      v2f a; a.x = p0; a.y = p1;                      // A 16x4: P[i][j]
      const float* h0 = hbase + (size_t)j0 * HID;
      const float* h1 = h0 + HID;
      v2f b0, b1;                                     // B 4x16: hw[j][f]
      b0.x = h0[r];       b0.y = h1[r];
      b1.x = h0[16 + r];  b1.y = h1[16 + r];
      c0 = __builtin_amdgcn_wmma_f32_16x16x4_f32(false, a, false, b0, (short)0, c0,
                                                 false, false);
      c1 = __builtin_amdgcn_wmma_f32_16x16x4_f32(false, a, false, b1, (short)0, c1,
                                                 false, false);
    }
  }
  float* op = out + ((size_t)b * GN) * HID + h * FH;
  for (int v = 0; v < 8; ++v) {
    int iw = it * 16 + v + 8 * half;                  // D: row = v + 8*half, col = r
    float v0 = c0[v], v1 = c1[v];
    if (relu) { v0 = fmaxf(v0, 0.0f); v1 = fmaxf(v1, 0.0f); }
    op[(size_t)iw * HID + r]      = v0;
    op[(size_t)iw * HID + 16 + r] = v1;
  }
}

// ---------------------------------------------------------------------------
extern "C" void kernel_launch(void* const* d_in, const int* in_sizes, int n_in,
                              void* d_out, int out_size, void* d_ws, size_t ws_size,
                              hipStream_t stream) {
  (void)in_sizes; (void)n_in; (void)out_size; (void)ws_size;
  const float* x      = (const float*)d_in[0];
  const float* emb    = (const float*)d_in[1];
  const float* proj_W = (const float*)d_in[2];
  const float* proj_b = (const float*)d_in[3];
  const float* W1     = (const float*)d_in[4];
  const float* a1_src = (const float*)d_in[5];
  const float* a1_dst = (const float*)d_in[6];
  const float* W2     = (const float*)d_in[7];
  const float* a2_src = (const float*)d_in[8];
  const float* a2_dst = (const float*)d_in[9];
  float* out = (float*)d_out;

  char* ws = (char*)d_ws;
  size_t off = 0;
  auto take = [&](size_t bytes) -> char* {
    char* p = ws + off;
    off += (bytes + 255) & ~(size_t)255;
    return p;
  };
  float*    ne    = (float*)take((size_t)GN * HID * sizeof(float));         // 1 MB
  unsigned* minS  = (unsigned*)take(256);
  unsigned* maskb = (unsigned*)take((size_t)GN * 64 * sizeof(unsigned));    // 512 KB
  float*    xp    = (float*)take((size_t)BB * GN * HID * sizeof(float));    // 8 MB (also h1)
  float*    hwb   = (float*)take((size_t)BB * GN * HID * sizeof(float));    // 8 MB
  float*    esrc  = (float*)take((size_t)BB * HEADS * GN * sizeof(float));
  float*    edst  = (float*)take((size_t)BB * HEADS * GN * sizeof(float));
  float*    mrow  = (float*)take((size_t)BB * HEADS * GN * sizeof(float));
  float*    zinv  = (float*)take((size_t)BB * HEADS * GN * sizeof(float));

  // graph build
  normalize_emb<<<GN, 128, 0, stream>>>(emb, ne, minS);
  row_top2_min<<<GN, 256, 0, stream>>>(ne, minS);
  build_mask<<<GN, 256, 0, stream>>>(ne, minS, maskb);

  const int M = BB * GN;                       // 16384 rows
  const int gemmGrid = (M / 16) * (HID / 16);  // 8192 waves

  // projection: xp = x @ proj_W + proj_b
  gemm_f32_wmma<<<gemmGrid, 32, 0, stream>>>(x, proj_W, proj_b, xp, M, INF_, HID, 0);

  // ---- GAT layer 1 ----
  gemm_f32_wmma<<<gemmGrid, 32, 0, stream>>>(xp, W1, nullptr, hwb, M, HID, HID, 0);
  edge_scores<<<(BB * HEADS * GN) / 256, 256, 0, stream>>>(hwb, a1_src, a1_dst, esrc, edst);
  attn_stats<<<BB * HEADS * GN, 256, 0, stream>>>(esrc, edst, maskb, mrow, zinv);
  attn_apply<<<BB * HEADS * (GN / 16), 32, 0, stream>>>(hwb, esrc, edst, mrow, zinv,
                                                        maskb, xp, 1);  // h1 (relu) -> xp

  // ---- GAT layer 2 ----
  gemm_f32_wmma<<<gemmGrid, 32, 0, stream>>>(xp, W2, nullptr, hwb, M, HID, HID, 0);
  edge_scores<<<(BB * HEADS * GN) / 256, 256, 0, stream>>>(hwb, a2_src, a2_dst, esrc, edst);
  attn_stats<<<BB * HEADS * GN, 256, 0, stream>>>(esrc, edst, maskb, mrow, zinv);
  attn_apply<<<BB * HEADS * (GN / 16), 32, 0, stream>>>(hwb, esrc, edst, mrow, zinv,
                                                        maskb, out, 0);
}